// CopyAttention_78632261255579
// MI455X (gfx1250) — compile-verified
//
#include <hip/hip_runtime.h>
#include <math.h>

typedef float v2f __attribute__((ext_vector_type(2)));
typedef float v8f __attribute__((ext_vector_type(8)));

// ---------------------------------------------------------------------------
// Problem constants (reference shapes are fixed).
// ---------------------------------------------------------------------------
#define BB 4
#define TT 512
#define SS 512
#define DD 1024
#define VV 32000
#define EE 512
#define LDOUT (VV + EE)       // 32512
#define ROWS (BB * TT)        // 2048

// Workspace layout (bytes)
#define WS_PCOPY_OFF 0u                         // ROWS floats
#define WS_QW_OFF    8192u                      // ROWS*DD floats (8 MB)
#define WS_ATTN_OFF  (8192u + 8388608u)         // BB*TT*SS floats (4 MB)

// ---------------------------------------------------------------------------
// p_copy = sigmoid(q . W_copy + b_copy), one block per row
// ---------------------------------------------------------------------------
__global__ __launch_bounds__(256) void pcopy_kernel(const float* __restrict__ q,
                                                    const float* __restrict__ Wc,
                                                    const float* __restrict__ bc,
                                                    float* __restrict__ pc) {
  const int row = blockIdx.x;
  const float* qr = q + (size_t)row * DD;
  float s = 0.f;
  for (int i = threadIdx.x; i < DD; i += 256) s += qr[i] * Wc[i];
  __shared__ float red[256];
  red[threadIdx.x] = s;
  __syncthreads();
  for (int off = 128; off > 0; off >>= 1) {
    if (threadIdx.x < off) red[threadIdx.x] += red[threadIdx.x + off];
    __syncthreads();
  }
  if (threadIdx.x == 0) {
    float z = red[0] + bc[0];
    pc[row] = 1.f / (1.f + expf(-z));
  }
}

// ---------------------------------------------------------------------------
// Generic fp32 WMMA GEMM: C = A @ B (+ bias), exact tiling required.
//   A: [M,K] row-major (lda), B: !BT -> [K,N] row-major (ldb)
//                             BT  -> logical B[k][n] stored at B[n*ldb + k]
//   C: [M,N] with row stride ldc. gridDim = (N/256, M/64, batches).
// Block: 256 threads = 8 waves (2 in M x 4 in N); wave tile 32x64 (2x4 WMMA
// subtiles). K staged 16-wide through double-buffered LDS with register
// staging so global loads overlap the 32 WMMAs per stage.
// ---------------------------------------------------------------------------
#define BLK_M 64
#define BLK_N 256
#define AS_STRIDE 20
#define BS_STRIDE 264

template <bool BT>
__global__ __launch_bounds__(256) void gemm_wmma(
    const float* __restrict__ A, int lda, long long strideA,
    const float* __restrict__ B, int ldb, long long strideB,
    float* __restrict__ C, int ldc, long long strideC,
    const float* __restrict__ bias, int K) {
  __shared__ float As[2][BLK_M * AS_STRIDE];
  __shared__ float Bs[2][16 * BS_STRIDE];

  const int tid  = threadIdx.x;
  const int lane = tid & 31;
  const int wave = tid >> 5;
  const int wm   = wave >> 2;   // 0..1  (M)
  const int wn   = wave & 3;    // 0..3  (N)
  const int l16  = lane & 15;
  const int lh   = lane >> 4;   // 0/1

  const int m0 = blockIdx.y * BLK_M;
  const int n0 = blockIdx.x * BLK_N;
  A += (size_t)blockIdx.z * strideA;
  B += (size_t)blockIdx.z * strideB;
  C += (size_t)blockIdx.z * strideC;

  // register staging indices
  const int ar  = tid >> 2;          // A row 0..63
  const int ac4 = (tid & 3) * 4;     // A col {0,4,8,12}

  float4 areg;
  float4 breg[4];

  auto load_global = [&](int k0) {
    areg = *(const float4*)(A + (size_t)(m0 + ar) * lda + k0 + ac4);
    if (!BT) {
      #pragma unroll
      for (int p = 0; p < 4; ++p) {
        const int idx = p * 256 + tid;     // 0..1023
        const int row = idx >> 6;          // 0..15
        const int c4  = (idx & 63) * 4;    // 0..252
        breg[p] = *(const float4*)(B + (size_t)(k0 + row) * ldb + n0 + c4);
      }
    } else {
      #pragma unroll
      for (int p = 0; p < 4; ++p) {
        const int idx = p * 256 + tid;
        const int nn  = idx >> 2;          // 0..255
        const int c4  = (idx & 3) * 4;     // 0..12
        breg[p] = *(const float4*)(B + (size_t)(n0 + nn) * ldb + k0 + c4);
      }
    }
  };

  auto store_lds = [&](int buf) {
    *(float4*)(&As[buf][ar * AS_STRIDE + ac4]) = areg;
    if (!BT) {
      #pragma unroll
      for (int p = 0; p < 4; ++p) {
        const int idx = p * 256 + tid;
        const int row = idx >> 6;
        const int c4  = (idx & 63) * 4;
        *(float4*)(&Bs[buf][row * BS_STRIDE + c4]) = breg[p];
      }
    } else {
      #pragma unroll
      for (int p = 0; p < 4; ++p) {
        const int idx = p * 256 + tid;
        const int nn  = idx >> 2;
        const int c4  = (idx & 3) * 4;
        Bs[buf][(c4 + 0) * BS_STRIDE + nn] = breg[p].x;
        Bs[buf][(c4 + 1) * BS_STRIDE + nn] = breg[p].y;
        Bs[buf][(c4 + 2) * BS_STRIDE + nn] = breg[p].z;
        Bs[buf][(c4 + 3) * BS_STRIDE + nn] = breg[p].w;
      }
    }
  };

  v8f acc[2][4] = {};

  const int nstage = K >> 4;
  load_global(0);
  store_lds(0);
  int cur = 0;

  for (int s = 0; s < nstage; ++s) {
    __syncthreads();                       // LDS buffer cur ready for all waves
    if (s + 1 < nstage) load_global((s + 1) << 4);   // prefetch into regs

    #pragma unroll
    for (int kk = 0; kk < 16; kk += 4) {
      v2f af[2], bf[4];
      #pragma unroll
      for (int i = 0; i < 2; ++i) {
        const int r = wm * 32 + i * 16 + l16;
        const int c = kk + lh * 2;
        af[i].x = As[cur][r * AS_STRIDE + c];
        af[i].y = As[cur][r * AS_STRIDE + c + 1];
      }
      #pragma unroll
      for (int j = 0; j < 4; ++j) {
        const int c = wn * 64 + j * 16 + l16;
        const int r = kk + lh * 2;
        bf[j].x = Bs[cur][r * BS_STRIDE + c];
        bf[j].y = Bs[cur][(r + 1) * BS_STRIDE + c];
      }
      #pragma unroll
      for (int i = 0; i < 2; ++i)
        #pragma unroll
        for (int j = 0; j < 4; ++j)
          acc[i][j] = __builtin_amdgcn_wmma_f32_16x16x4_f32(
              false, af[i], false, bf[j], (short)0, acc[i][j], false, false);
    }

    if (s + 1 < nstage) store_lds(cur ^ 1);  // other buffer: no WAR with cur
    cur ^= 1;
  }

  // ---- epilogue: C/D layout -> VGPR r holds rows r (lanes 0-15) / r+8 ----
  #pragma unroll
  for (int i = 0; i < 2; ++i) {
    #pragma unroll
    for (int j = 0; j < 4; ++j) {
      const int n = n0 + wn * 64 + j * 16 + l16;
      const float bb = bias ? bias[n] : 0.f;
      #pragma unroll
      for (int r = 0; r < 8; ++r) {
        const int m = m0 + wm * 32 + i * 16 + lh * 8 + r;
        C[(size_t)m * ldc + n] = acc[i][j][r] + bb;
      }
    }
  }
}

// ---------------------------------------------------------------------------
// In-place softmax over the V entries of each output row, times (1 - p_copy).
// PAD column (j==0) is treated as logit -1e-20, matching the reference.
// ---------------------------------------------------------------------------
__global__ __launch_bounds__(256) void softmax_gen(float* __restrict__ out,
                                                   const float* __restrict__ pc) {
  const int row = blockIdx.x;
  float* p = out + (size_t)row * LDOUT;
  const int tid = threadIdx.x;
  __shared__ float red[256];

  float mx = -3.0e38f;
  for (int j = tid; j < VV; j += 256) {
    const float v = (j == 0) ? -1e-20f : p[j];
    mx = fmaxf(mx, v);
  }
  red[tid] = mx;
  __syncthreads();
  for (int off = 128; off > 0; off >>= 1) {
    if (tid < off) red[tid] = fmaxf(red[tid], red[tid + off]);
    __syncthreads();
  }
  mx = red[0];
  __syncthreads();

  float sum = 0.f;
  for (int j = tid; j < VV; j += 256) {
    const float v = (j == 0) ? -1e-20f : p[j];
    sum += expf(v - mx);
  }
  red[tid] = sum;
  __syncthreads();
  for (int off = 128; off > 0; off >>= 1) {
    if (tid < off) red[tid] += red[tid + off];
    __syncthreads();
  }
  const float scale = (1.f - pc[row]) / red[0];

  for (int j = tid; j < VV; j += 256) {
    const float v = (j == 0) ? -1e-20f : p[j];
    p[j] = expf(v - mx) * scale;
  }
}

// ---------------------------------------------------------------------------
// In-place masked softmax over S for each (b,t) row of attn, times p_copy.
// ---------------------------------------------------------------------------
__global__ __launch_bounds__(256) void softmax_copy(float* __restrict__ attn,
                                                    const float* __restrict__ pc,
                                                    const int* __restrict__ mask) {
  const int row = blockIdx.x;            // b*T + t
  const int b = row / TT;
  float* p = attn + (size_t)row * SS;
  const int* mrow = mask + (size_t)b * SS;
  const int tid = threadIdx.x;
  __shared__ float red[256];

  float mx = -3.0e38f;
  for (int j = tid; j < SS; j += 256) {
    const float v = mrow[j] ? -1e18f : p[j];
    mx = fmaxf(mx, v);
  }
  red[tid] = mx;
  __syncthreads();
  for (int off = 128; off > 0; off >>= 1) {
    if (tid < off) red[tid] = fmaxf(red[tid], red[tid + off]);
    __syncthreads();
  }
  mx = red[0];
  __syncthreads();

  float sum = 0.f;
  for (int j = tid; j < SS; j += 256) {
    const float v = mrow[j] ? -1e18f : p[j];
    sum += expf(v - mx);
  }
  red[tid] = sum;
  __syncthreads();
  for (int off = 128; off > 0; off >>= 1) {
    if (tid < off) red[tid] += red[tid + off];
    __syncthreads();
  }
  const float scale = pc[row] / red[0];

  for (int j = tid; j < SS; j += 256) {
    const float v = mrow[j] ? -1e18f : p[j];
    p[j] = expf(v - mx) * scale;
  }
}

// ---------------------------------------------------------------------------
// Launch
// ---------------------------------------------------------------------------
extern "C" void kernel_launch(void* const* d_in, const int* in_sizes, int n_in,
                              void* d_out, int out_size, void* d_ws, size_t ws_size,
                              hipStream_t stream) {
  const float* query  = (const float*)d_in[0];   // [B,T,D]
  const float* mbank  = (const float*)d_in[1];   // [B,S,D]
  const int*   mask   = (const int*)d_in[2];     // [B,S] bool -> int32
  const float* srcmap = (const float*)d_in[3];   // [B,S,E]
  const float* W_in   = (const float*)d_in[4];   // [D,D]
  const float* W_copy = (const float*)d_in[5];   // [D,1]
  const float* b_copy = (const float*)d_in[6];   // [1]
  const float* W_gen  = (const float*)d_in[7];   // [D,V]
  const float* b_gen  = (const float*)d_in[8];   // [V]
  float* out = (float*)d_out;                    // [B,T,V+E]

  char* ws = (char*)d_ws;
  float* pc   = (float*)(ws + WS_PCOPY_OFF);     // [ROWS]
  float* qw   = (float*)(ws + WS_QW_OFF);        // [ROWS, D]
  float* attn = (float*)(ws + WS_ATTN_OFF);      // [B, T, S]

  // 1) copy gate
  pcopy_kernel<<<ROWS, 256, 0, stream>>>(query, W_copy, b_copy, pc);

  // 2) logits = Q @ W_gen + b_gen, written into out[..., 0:V] (strided)
  gemm_wmma<false><<<dim3(VV / BLK_N, ROWS / BLK_M, 1), 256, 0, stream>>>(
      query, DD, 0, W_gen, VV, 0, out, LDOUT, 0, b_gen, DD);

  // 3) softmax over V, * (1 - p_copy), in place
  softmax_gen<<<ROWS, 256, 0, stream>>>(out, pc);

  // 4) QW = Q @ W_in
  gemm_wmma<false><<<dim3(DD / BLK_N, ROWS / BLK_M, 1), 256, 0, stream>>>(
      query, DD, 0, W_in, DD, 0, qw, DD, 0, nullptr, DD);

  // 5) attn[b] = QW[b] @ memory_bank[b]^T   (batched, B transposed)
  gemm_wmma<true><<<dim3(SS / BLK_N, TT / BLK_M, BB), 256, 0, stream>>>(
      qw, DD, (long long)TT * DD, mbank, DD, (long long)SS * DD,
      attn, SS, (long long)TT * SS, nullptr, DD);

  // 6) masked softmax over S, * p_copy, in place
  softmax_copy<<<ROWS, 256, 0, stream>>>(attn, pc, mask);

  // 7) copy_prob[b] = attn[b] @ src_map[b], into out[..., V:V+E]
  gemm_wmma<false><<<dim3(EE / BLK_N, TT / BLK_M, BB), 256, 0, stream>>>(
      attn, SS, (long long)TT * SS, srcmap, EE, (long long)SS * EE,
      out + VV, LDOUT, (long long)TT * LDOUT, nullptr, SS);
}